// TrilinearInterpolateSQN_26225070309541
// MI455X (gfx1250) — compile-verified
//
#include <hip/hip_runtime.h>

// Problem constants (match reference setup_inputs)
#define BATCH   8
#define MKNOWN  2048
#define CFEAT   128
#define EPSV    1e-8f
#define FLT_BIG 3.402823466e+38f

typedef __attribute__((ext_vector_type(2))) float v2f;
typedef __attribute__((ext_vector_type(8))) float v8f;
typedef __attribute__((ext_vector_type(4))) int   v4i;

#if __has_builtin(__builtin_amdgcn_wmma_f32_16x16x4_f32)
#define USE_WMMA_F32X4 1
#else
#define USE_WMMA_F32X4 0
#endif

#if __has_builtin(__builtin_amdgcn_global_load_async_to_lds_b128) && \
    __has_builtin(__builtin_amdgcn_s_wait_asynccnt)
#define USE_ASYNC_LDS 1
#else
#define USE_ASYNC_LDS 0
#endif

#if USE_ASYNC_LDS
typedef __attribute__((address_space(1))) v4i gv4i;   // global int4
typedef __attribute__((address_space(3))) v4i lv4i;   // LDS int4

__device__ __forceinline__ void async_g2l_b128(const void* g, void* l) {
    __builtin_amdgcn_global_load_async_to_lds_b128((gv4i*)g, (lv4i*)l, 0, 0);
}
#endif

// Branchless sorted top-3 insert.
// Distances: pure min/max network (no conditions).
// Indices: single-phi "if (c) x = y;" diamonds -> folded to v_cndmask.
__device__ __forceinline__ void top3_insert(float v, int col,
                                            float& t0, float& t1, float& t2,
                                            int& i0, int& i1, int& i2) {
    const float o0 = t0, o1 = t1, o2 = t2;
    t0 = fminf(o0, v);
    t1 = fminf(fmaxf(v, o0), o1);
    t2 = fminf(fmaxf(v, o1), o2);
    const bool c0 = v < o0;
    const bool c1 = v < o1;
    const bool c2 = v < o2;
    int m0 = i0; if (c0) m0 = col;
    int m1 = i1; if (c1) m1 = col;
    if (c0) m1 = i0;
    int m2 = i2; if (c2) m2 = col;
    if (c1) m2 = i1;
    i0 = m0; i1 = m1; i2 = m2;
}

// ---------------- workspace layout (ints) ----------------
// ws[0..7]   counts per batch
// ws[8..15]  scatter cursors per batch
// ws[16..24] offsets (16-aligned segment starts), offsets[8] = padded total
// ws[32..]   perm array: n + BATCH*16 slots, -1 = padding

__global__ void init_ws_k(int* __restrict__ ws, int P) {
    int g = blockIdx.x * blockDim.x + threadIdx.x;
    if (g < 16) ws[g] = 0;
    if (g < P)  ws[32 + g] = -1;
}

__global__ void count_k(const int* __restrict__ bi, int* __restrict__ ws, int n) {
    int g = blockIdx.x * blockDim.x + threadIdx.x;
    if (g < n) atomicAdd(&ws[bi[g]], 1);
}

__global__ void prefix_k(int* __restrict__ ws) {
    int* offs = ws + 16;
    offs[0] = 0;
    for (int b = 0; b < BATCH; ++b)
        offs[b + 1] = offs[b] + ((ws[b] + 15) & ~15);
}

__global__ void scatter_k(const int* __restrict__ bi, int* __restrict__ ws, int n) {
    int g = blockIdx.x * blockDim.x + threadIdx.x;
    if (g < n) {
        int b = bi[g];
        int pos = atomicAdd(&ws[8 + b], 1);
        ws[32 + ws[16 + b] + pos] = g;
    }
}

// ---------------- main kernel: one wave per 16-query tile ----------------
__global__ __launch_bounds__(32) void interp_k(
    const float* __restrict__ unknown, const float* __restrict__ known,
    const float* __restrict__ feats, const int* __restrict__ ws,
    float* __restrict__ out)
{
    __shared__ float s_kn[MKNOWN * 3];   // 24KB: this batch's known coords
    __shared__ float s_cd[32 * 8 * 3];   // per-lane top-3 distances
    __shared__ int   s_ci[32 * 8 * 3];   // per-lane top-3 indices
    __shared__ float s_w[16 * 3];
    __shared__ int   s_ix[16 * 3];
    __shared__ int   s_row[16];
    __shared__ float s_usq[16];
#if !USE_WMMA_F32X4
    __shared__ float s_u[16 * 3];
#endif

    const int lane   = threadIdx.x;
    const int tstart = blockIdx.x * 16;
    const int* offs  = ws + 16;
    if (tstart >= offs[BATCH]) return;   // uniform exit: EXEC stays all-1 for WMMA

    int b = 0;
    #pragma unroll
    for (int bb = 0; bb < BATCH - 1; ++bb)
        if (tstart >= offs[bb + 1]) b = bb + 1;

    // Kick off async staging of this batch's known coords into LDS first,
    // so the DMA overlaps the query-side setup below.
    {
        const float4* kb4 = (const float4*)(known + (size_t)b * MKNOWN * 3);
        float4* s4 = (float4*)s_kn;
#if USE_ASYNC_LDS
        #pragma unroll
        for (int t = 0; t < (MKNOWN * 3) / 4 / 32; ++t)   // 48 iters/lane
            async_g2l_b128(kb4 + (t * 32 + lane), s4 + (t * 32 + lane));
#else
        #pragma unroll 4
        for (int t = lane; t < (MKNOWN * 3) / 4; t += 32) s4[t] = kb4[t];
#endif
    }

    const int* perm = ws + 32;
    const int myrow = perm[tstart + (lane & 15)];   // row M = lane%16 of this tile

    float ux = 0.f, uy = 0.f, uz = 0.f;
    if (myrow >= 0) {
        ux = unknown[myrow * 3 + 0];
        uy = unknown[myrow * 3 + 1];
        uz = unknown[myrow * 3 + 2];
    }
    if (lane < 16) {
        s_row[lane] = myrow;
        s_usq[lane] = ux * ux + uy * uy + uz * uz;
#if !USE_WMMA_F32X4
        s_u[lane * 3 + 0] = ux; s_u[lane * 3 + 1] = uy; s_u[lane * 3 + 2] = uz;
#endif
    }

    // A fragment (16x4 f32): lanes 0-15 hold K=0,1 ; lanes 16-31 hold K=2,3
    // A row = (-2ux, -2uy, -2uz, 1)
    v2f afrag;
    afrag[0] = (lane < 16) ? (-2.f * ux) : (-2.f * uz);
    afrag[1] = (lane < 16) ? (-2.f * uy) : 1.0f;

#if USE_ASYNC_LDS
    __builtin_amdgcn_s_wait_asynccnt(0);   // LDS tile landed
#endif
    __syncthreads();

    // C accumulator = |u|^2 broadcast along each row -> D is exact d^2
    v8f cfrag;
    {
        const int rbase = (lane < 16) ? 0 : 8;
        #pragma unroll
        for (int r = 0; r < 8; ++r) cfrag[r] = s_usq[rbase + r];
    }

    // Running per-lane top-3 for 8 (row, column-stream) pairs
    float t0[8], t1[8], t2[8];
    int   i0[8], i1[8], i2[8];
    #pragma unroll
    for (int r = 0; r < 8; ++r) {
        t0[r] = t1[r] = t2[r] = FLT_BIG;
        i0[r] = i1[r] = i2[r] = 0;
    }

    const int colres = lane & 15;
    for (int j0 = 0; j0 < MKNOWN; j0 += 16) {
        const int col = j0 + colres;
        const float kx = s_kn[col * 3 + 0];
        const float ky = s_kn[col * 3 + 1];
        const float kz = s_kn[col * 3 + 2];
        // B fragment (4x16 f32): lanes 0-15 K=0,1 ; lanes 16-31 K=2,3
        // B col = (kx, ky, kz, |k|^2)
        v2f bfrag;
        bfrag[0] = (lane < 16) ? kx : kz;
        bfrag[1] = (lane < 16) ? ky : (kx * kx + ky * ky + kz * kz);

        v8f d;
#if USE_WMMA_F32X4
        d = __builtin_amdgcn_wmma_f32_16x16x4_f32(
                false, afrag, false, bfrag, (short)0, cfrag, false, false);
#else
        #pragma unroll
        for (int r = 0; r < 8; ++r) {
            const int row = ((lane < 16) ? 0 : 8) + r;
            const float dx = s_u[row * 3 + 0] - kx;
            const float dy = s_u[row * 3 + 1] - ky;
            const float dz = s_u[row * 3 + 2] - kz;
            d[r] = dx * dx + dy * dy + dz * dz;
        }
#endif
        #pragma unroll
        for (int r = 0; r < 8; ++r)
            top3_insert(d[r], col, t0[r], t1[r], t2[r], i0[r], i1[r], i2[r]);
    }

    // Dump per-lane candidates to LDS for cross-lane merge
    #pragma unroll
    for (int r = 0; r < 8; ++r) {
        const int base = (lane * 8 + r) * 3;
        s_cd[base + 0] = t0[r]; s_cd[base + 1] = t1[r]; s_cd[base + 2] = t2[r];
        s_ci[base + 0] = i0[r]; s_ci[base + 1] = i1[r]; s_ci[base + 2] = i2[r];
    }
    __syncthreads();

    // Lane R (0..15) owns row R: merge 16 lanes x 3 candidates -> global top-3
    if (lane < 16) {
        float b0 = FLT_BIG, b1 = FLT_BIG, b2 = FLT_BIG;
        int   j0_ = 0, j1_ = 0, j2_ = 0;
        const int rr    = lane & 7;
        const int lbase = (lane < 8) ? 0 : 16;
        for (int l = 0; l < 16; ++l) {
            const int base = ((lbase + l) * 8 + rr) * 3;
            #pragma unroll
            for (int k = 0; k < 3; ++k)
                top3_insert(s_cd[base + k], s_ci[base + k], b0, b1, b2, j0_, j1_, j2_);
        }
        const float d0 = sqrtf(fmaxf(b0, 0.f));
        const float d1 = sqrtf(fmaxf(b1, 0.f));
        const float d2 = sqrtf(fmaxf(b2, 0.f));
        const float r0 = 1.f / (d0 + EPSV);
        const float r1 = 1.f / (d1 + EPSV);
        const float r2 = 1.f / (d2 + EPSV);
        const float rs = 1.f / (r0 + r1 + r2);
        s_w[lane * 3 + 0] = r0 * rs; s_w[lane * 3 + 1] = r1 * rs; s_w[lane * 3 + 2] = r2 * rs;
        s_ix[lane * 3 + 0] = j0_;    s_ix[lane * 3 + 1] = j1_;    s_ix[lane * 3 + 2] = j2_;
    }
    __syncthreads();

    // Feature gather/blend: whole wave sweeps C=128 channels per row
    const float* fb = feats + (size_t)b * CFEAT * MKNOWN;
    for (int R = 0; R < 16; ++R) {
        const int urow = s_row[R];
        if (urow < 0) continue;
        const float w0 = s_w[R * 3 + 0], w1 = s_w[R * 3 + 1], w2 = s_w[R * 3 + 2];
        const int   a0 = s_ix[R * 3 + 0], a1 = s_ix[R * 3 + 1], a2 = s_ix[R * 3 + 2];
        #pragma unroll
        for (int ci = 0; ci < CFEAT / 32; ++ci) {
            const int c = ci * 32 + lane;
            const float* fc = fb + (size_t)c * MKNOWN;
            out[(size_t)urow * CFEAT + c] = w0 * fc[a0] + w1 * fc[a1] + w2 * fc[a2];
        }
    }
}

extern "C" void kernel_launch(void* const* d_in, const int* in_sizes, int n_in,
                              void* d_out, int out_size, void* d_ws, size_t ws_size,
                              hipStream_t stream) {
    (void)n_in; (void)out_size; (void)ws_size;
    const float* unknown = (const float*)d_in[0];
    const float* known   = (const float*)d_in[1];
    const int*   binds   = (const int*)d_in[2];
    const float* feats   = (const float*)d_in[3];
    float* out = (float*)d_out;
    int*   ws  = (int*)d_ws;

    const int n = in_sizes[2];          // number of unknown points
    const int P = n + BATCH * 16;       // perm slots incl. worst-case padding

    init_ws_k<<<(P + 255) / 256, 256, 0, stream>>>(ws, P);
    count_k  <<<(n + 255) / 256, 256, 0, stream>>>(binds, ws, n);
    prefix_k <<<1, 1, 0, stream>>>(ws);
    scatter_k<<<(n + 255) / 256, 256, 0, stream>>>(binds, ws, n);

    const int tiles = (n + BATCH * 15 + 15) / 16;   // upper bound on padded tiles
    interp_k<<<tiles, 32, 0, stream>>>(unknown, known, feats, ws, out);
}